// GraphNeuralNetwork_54133767798874
// MI455X (gfx1250) — compile-verified
//
#include <hip/hip_runtime.h>

typedef __attribute__((ext_vector_type(2))) float v2f;
typedef __attribute__((ext_vector_type(8))) float v8f;

#define NNODES 100000
#define NEDGES 1600000
#define NGRAPH 4096
#define FIN    11
#define HD     128
#define HD2    64
#define NLAYER 3

// ---------------------------------------------------------------- utilities
__global__ void fzero_kernel(float* __restrict__ p, int n) {
    int i = blockIdx.x * blockDim.x + threadIdx.x;
    if (i < n) p[i] = 0.0f;
}

// Pack W[K][NC] row-major -> Wp[K/4][NC][4] so that the two k-values a WMMA
// lane needs (k+kp, k+kp+1) are contiguous (one b64 load, 8B aligned).
__global__ void pack_w_kernel(const float* __restrict__ W, float* __restrict__ Wp,
                              int K, int NC) {
    int idx = blockIdx.x * blockDim.x + threadIdx.x;
    if (idx >= K * NC) return;
    int k = idx / NC;
    int c = idx - k * NC;
    Wp[((size_t)(k >> 2) * NC + c) * 4 + (k & 3)] = W[idx];
}

// ------------------------------------------------- embedding: relu(x@W + b)
__global__ void embed_relu_kernel(const float* __restrict__ x,
                                  const float* __restrict__ W,
                                  const float* __restrict__ b,
                                  float* __restrict__ h) {
    int idx = blockIdx.x * blockDim.x + threadIdx.x;
    if (idx >= NNODES * HD) return;
    int nrow = idx >> 7;
    int j    = idx & (HD - 1);
    const float* xr = x + (size_t)nrow * FIN;
    float acc = b[j];
#pragma unroll
    for (int k = 0; k < FIN; ++k) acc = fmaf(xr[k], W[k * HD + j], acc);
    h[idx] = fmaxf(acc, 0.0f);
}

// ------------------------------------- edge scatter: msg[dst] += h[src]
// one wave (32 lanes) per edge, float4 per lane (128 features);
// prefetch the gather row of a future edge to hide random-access latency.
#define PF_DIST 8
__global__ void scatter_add_kernel(const int* __restrict__ src,
                                   const int* __restrict__ dst,
                                   const float* __restrict__ h,
                                   float* __restrict__ msg) {
    int gid  = blockIdx.x * blockDim.x + threadIdx.x;
    int e    = gid >> 5;
    int lane = threadIdx.x & 31;
    if (e >= NEDGES) return;
    if (e + PF_DIST * 8 < NEDGES) {  // wave-uniform guard
        int sp = src[e + PF_DIST * 8];
        __builtin_prefetch(h + (size_t)sp * HD + lane * 4, 0, 1);
    }
    int s = src[e];
    int d = dst[e];
    const float4 v = *(const float4*)(h + (size_t)s * HD + lane * 4);
    float* p = msg + (size_t)d * HD + lane * 4;
    __hip_atomic_fetch_add(p + 0, v.x, __ATOMIC_RELAXED, __HIP_MEMORY_SCOPE_AGENT);
    __hip_atomic_fetch_add(p + 1, v.y, __ATOMIC_RELAXED, __HIP_MEMORY_SCOPE_AGENT);
    __hip_atomic_fetch_add(p + 2, v.z, __ATOMIC_RELAXED, __HIP_MEMORY_SCOPE_AGENT);
    __hip_atomic_fetch_add(p + 3, v.w, __ATOMIC_RELAXED, __HIP_MEMORY_SCOPE_AGENT);
}

// ---------------------------------------------------------------- WMMA GEMM
// out[M,NC] = relu( A@W (+ A2@W2) + bias ), K = 128, fp32 WMMA 16x16x4.
// W given in packed [K/4][NC][4] layout (see pack_w_kernel).
// blockDim.x = 32*(NC/16): wave ct = column tile. Each wave computes MT
// consecutive 16-row tiles sharing B fragments (4x B reuse, 4 indep. chains).
// fp32 WMMA layouts (ISA 7.12.2):
//   A 16x4 : lane<16 -> M=lane, K={0,1}; lane>=16 -> M=lane-16, K={2,3}
//   B 4x16 : lane<16 -> N=lane, K={0,1}; lane>=16 -> N=lane-16, K={2,3}
//   C/D    : VGPR i, lane<16 -> M=i, N=lane; lane>=16 -> M=i+8, N=lane-16
template <int NC, int MT>
__global__ void gemm_wmma_f32_kernel(const float* __restrict__ A,
                                     const float* __restrict__ Wp,
                                     const float* __restrict__ A2,
                                     const float* __restrict__ Wp2,
                                     const float* __restrict__ bias,
                                     float* __restrict__ out,
                                     int Mtiles) {
    constexpr int K = HD;
    const int lane = threadIdx.x & 31;
    const int ct   = threadIdx.x >> 5;
    const int rt0  = blockIdx.x * MT;
    const int m    = lane & 15;
    const int half = lane >> 4;      // 0 or 1
    const int kp   = half << 1;      // 0 or 2
    const int col  = ct * 16 + m;

    v8f acc[MT];
    const float* arow[MT];
    const float* arow2[MT];
    bool valid[MT];  // uniform across wave (depends only on blockIdx)
#pragma unroll
    for (int t = 0; t < MT; ++t) {
        acc[t] = (v8f){};
        valid[t] = (rt0 + t) < Mtiles;
        int r = valid[t] ? (rt0 + t) * 16 + m : m;  // clamp to safe rows
        arow[t]  = A + (size_t)r * K;
        arow2[t] = (A2 != nullptr) ? (A2 + (size_t)r * K) : nullptr;
    }

#pragma unroll 2
    for (int kg = 0; kg < K / 4; ++kg) {
        const int k = kg * 4;
        v2f b;
        {
            const float* wp = Wp + ((size_t)kg * NC + col) * 4 + kp;
            b.x = wp[0];
            b.y = wp[1];
        }
#pragma unroll
        for (int t = 0; t < MT; ++t) {
            v2f a;
            a.x = arow[t][k + kp];
            a.y = arow[t][k + kp + 1];
            acc[t] = __builtin_amdgcn_wmma_f32_16x16x4_f32(
                false, a, false, b, (short)0, acc[t], false, false);
        }
        if (A2 != nullptr) {  // uniform branch
            v2f b2;
            const float* wp2 = Wp2 + ((size_t)kg * NC + col) * 4 + kp;
            b2.x = wp2[0];
            b2.y = wp2[1];
#pragma unroll
            for (int t = 0; t < MT; ++t) {
                v2f a;
                a.x = arow2[t][k + kp];
                a.y = arow2[t][k + kp + 1];
                acc[t] = __builtin_amdgcn_wmma_f32_16x16x4_f32(
                    false, a, false, b2, (short)0, acc[t], false, false);
            }
        }
    }

    const float bv   = bias[col];
    const int   rofs = half << 3;  // 0 or 8
#pragma unroll
    for (int t = 0; t < MT; ++t) {
        if (!valid[t]) continue;  // uniform
        const int rbase = (rt0 + t) * 16 + rofs;
#pragma unroll
        for (int i = 0; i < 8; ++i) {
            float v = acc[t][i] + bv;
            v = fmaxf(v, 0.0f);
            out[(size_t)(rbase + i) * NC + col] = v;
        }
    }
}

// -------------------------------- pooling: pooled[batch[n]] += h[n]; counts
__global__ void pool_scatter_kernel(const int* __restrict__ batch,
                                    const float* __restrict__ h,
                                    float* __restrict__ pooled,
                                    float* __restrict__ counts) {
    int gid  = blockIdx.x * blockDim.x + threadIdx.x;
    int node = gid >> 5;
    int lane = threadIdx.x & 31;
    if (node >= NNODES) return;
    int g = batch[node];
    const float4 v = *(const float4*)(h + (size_t)node * HD + lane * 4);
    float* p = pooled + (size_t)g * HD + lane * 4;
    __hip_atomic_fetch_add(p + 0, v.x, __ATOMIC_RELAXED, __HIP_MEMORY_SCOPE_AGENT);
    __hip_atomic_fetch_add(p + 1, v.y, __ATOMIC_RELAXED, __HIP_MEMORY_SCOPE_AGENT);
    __hip_atomic_fetch_add(p + 2, v.z, __ATOMIC_RELAXED, __HIP_MEMORY_SCOPE_AGENT);
    __hip_atomic_fetch_add(p + 3, v.w, __ATOMIC_RELAXED, __HIP_MEMORY_SCOPE_AGENT);
    if (lane == 0)
        __hip_atomic_fetch_add(counts + g, 1.0f, __ATOMIC_RELAXED, __HIP_MEMORY_SCOPE_AGENT);
}

__global__ void pool_divide_kernel(float* __restrict__ pooled,
                                   const float* __restrict__ counts) {
    int i = blockIdx.x * blockDim.x + threadIdx.x;
    if (i >= NGRAPH * HD) return;
    pooled[i] = pooled[i] / fmaxf(counts[i >> 7], 1.0f);
}

// ------------------------------------------ final head: out = m2@W3 + b3
__global__ void head3_kernel(const float* __restrict__ m2,
                             const float* __restrict__ W3,
                             const float* __restrict__ b3,
                             float* __restrict__ out) {
    int gid  = blockIdx.x * blockDim.x + threadIdx.x;
    int g    = gid >> 5;
    int lane = threadIdx.x & 31;
    if (g >= NGRAPH) return;
    const float* row = m2 + (size_t)g * HD2;
    float s = fmaf(row[lane], W3[lane], row[32 + lane] * W3[32 + lane]);
#pragma unroll
    for (int off = 16; off > 0; off >>= 1) s += __shfl_down(s, off, 32);
    if (lane == 0) out[g] = s + b3[0];
}

// --------------------------------------------------------------------------
extern "C" void kernel_launch(void* const* d_in, const int* in_sizes, int n_in,
                              void* d_out, int out_size, void* d_ws, size_t ws_size,
                              hipStream_t stream) {
    const float* x      = (const float*)d_in[0];
    const int*   eidx   = (const int*)d_in[1];
    const int*   batch  = (const int*)d_in[2];
    const float* W_emb  = (const float*)d_in[3];
    const float* b_emb  = (const float*)d_in[4];
    const float* W_rel  = (const float*)d_in[5];
    const float* W_root = (const float*)d_in[6];
    const float* b_conv = (const float*)d_in[7];
    const float* W1     = (const float*)d_in[8];
    const float* b1     = (const float*)d_in[9];
    const float* W2     = (const float*)d_in[10];
    const float* b2     = (const float*)d_in[11];
    const float* W3     = (const float*)d_in[12];
    const float* b3     = (const float*)d_in[13];
    float* out = (float*)d_out;

    const int* src = eidx;            // edge_index[0]
    const int* dst = eidx + NEDGES;   // edge_index[1]

    // workspace carve (floats)
    float* ws = (float*)d_ws;
    const size_t NH = (size_t)NNODES * HD;   // 12.8M
    float* hA      = ws;  ws += NH;
    float* hB      = ws;  ws += NH;
    float* msg     = ws;  ws += NH;
    float* pooled  = ws;  ws += (size_t)NGRAPH * HD;
    float* counts  = ws;  ws += NGRAPH;
    float* m1      = ws;  ws += (size_t)NGRAPH * HD;
    float* m2      = ws;  ws += (size_t)NGRAPH * HD2;
    float* wp_rel  = ws;  ws += (size_t)NLAYER * HD * HD;
    float* wp_root = ws;  ws += (size_t)NLAYER * HD * HD;
    float* wp1     = ws;  ws += (size_t)HD * HD;
    float* wp2     = ws;  ws += (size_t)HD * HD2;

    // 0) pack all WMMA weight matrices
    {
        const int g128 = (HD * HD + 255) / 256;
        for (int l = 0; l < NLAYER; ++l) {
            pack_w_kernel<<<g128, 256, 0, stream>>>(
                W_rel + (size_t)l * HD * HD, wp_rel + (size_t)l * HD * HD, HD, HD);
            pack_w_kernel<<<g128, 256, 0, stream>>>(
                W_root + (size_t)l * HD * HD, wp_root + (size_t)l * HD * HD, HD, HD);
        }
        pack_w_kernel<<<g128, 256, 0, stream>>>(W1, wp1, HD, HD);
        pack_w_kernel<<<(HD * HD2 + 255) / 256, 256, 0, stream>>>(W2, wp2, HD, HD2);
    }

    // 1) embedding
    embed_relu_kernel<<<(NNODES * HD + 255) / 256, 256, 0, stream>>>(x, W_emb, b_emb, hA);

    // 2) GraphConv layers
    constexpr int MT = 4;
    const int mtiles = NNODES / 16;                 // 6250
    const int gblk   = (mtiles + MT - 1) / MT;      // 1563
    float* hcur  = hA;
    float* hnext = hB;
    for (int l = 0; l < NLAYER; ++l) {
        fzero_kernel<<<(int)((NH + 255) / 256), 256, 0, stream>>>(msg, (int)NH);
        scatter_add_kernel<<<(NEDGES * 32) / 256, 256, 0, stream>>>(src, dst, hcur, msg);
        gemm_wmma_f32_kernel<HD, MT><<<gblk, 32 * (HD / 16), 0, stream>>>(
            msg, wp_rel + (size_t)l * HD * HD,
            hcur, wp_root + (size_t)l * HD * HD,
            b_conv + (size_t)l * HD, hnext, mtiles);
        float* t = hcur; hcur = hnext; hnext = t;
    }

    // 3) global mean pool
    fzero_kernel<<<(NGRAPH * HD + NGRAPH + 255) / 256, 256, 0, stream>>>(
        pooled, NGRAPH * HD + NGRAPH);   // pooled and counts are contiguous
    pool_scatter_kernel<<<(NNODES * 32 + 255) / 256, 256, 0, stream>>>(
        batch, hcur, pooled, counts);
    pool_divide_kernel<<<(NGRAPH * HD + 255) / 256, 256, 0, stream>>>(pooled, counts);

    // 4) MLP head
    const int gtiles = NGRAPH / 16;  // 256
    gemm_wmma_f32_kernel<HD, MT><<<gtiles / MT, 32 * (HD / 16), 0, stream>>>(
        pooled, wp1, nullptr, nullptr, b1, m1, gtiles);
    gemm_wmma_f32_kernel<HD2, MT><<<gtiles / MT, 32 * (HD2 / 16), 0, stream>>>(
        m1, wp2, nullptr, nullptr, b2, m2, gtiles);
    head3_kernel<<<(NGRAPH * 32) / 256, 256, 0, stream>>>(m2, W3, b3, out);
}